// FNN3d_10247791968397
// MI455X (gfx1250) — compile-verified
//
#include <hip/hip_runtime.h>
#include <hip/hip_bf16.h>
#include <math.h>

typedef __attribute__((ext_vector_type(2))) float v2f;
typedef __attribute__((ext_vector_type(8))) float v8f;

static constexpr int BATCH = 4;
static constexpr int GX = 128, GY = 128, GT = 20;
static constexpr int S  = GX * GY * GT;   // 327680 points per (b, channel)
static constexpr int XY = GX * GY;        // 16384
static constexpr float TWO_PI = 6.28318530717958647692f;

__device__ __forceinline__ float gelu_exact(float v) {
    return 0.5f * v * (1.0f + erff(v * 0.70710678118654752f));
}

// ---------------------------------------------------------------------------
// fc0: x (B,X,Y,T,5) @ w(5,20) + b -> h (B,20,X,Y,T)  channels-first
// ---------------------------------------------------------------------------
__global__ void k_fc0(const float* __restrict__ x, const float* __restrict__ w,
                      const float* __restrict__ bias, float* __restrict__ h, int total) {
    int idx = blockIdx.x * blockDim.x + threadIdx.x;
    if (idx >= total) return;
    int b = idx / S, pt = idx % S;
    float xi[5];
#pragma unroll
    for (int i = 0; i < 5; ++i) xi[i] = x[idx * 5 + i];
#pragma unroll
    for (int o = 0; o < 20; ++o) {
        float acc = bias[o];
#pragma unroll
        for (int i = 0; i < 5; ++i) acc += xi[i] * w[i * 20 + o];
        h[(b * 20 + o) * S + pt] = acc;
    }
}

// ---------------------------------------------------------------------------
// Forward partial DFT along T: real(20) -> complex(8 bins).
// rows r = (b*Ci+c)*XY + xy ;  ft1[r*16 + k] = Re, ft1[r*16+8+k] = Im
// ---------------------------------------------------------------------------
__global__ void k_dft_t(const float* __restrict__ h, float* __restrict__ ft1, int total) {
    __shared__ float tc[GT * 8], ts[GT * 8];
    for (int i = threadIdx.x; i < GT * 8; i += blockDim.x) {
        int t = i >> 3, k = i & 7;
        float ang = -TWO_PI * (float)(k * t) / (float)GT;
        float s, c; __sincosf(ang, &s, &c);
        tc[i] = c; ts[i] = s;
    }
    __syncthreads();
    int r = blockIdx.x * blockDim.x + threadIdx.x;
    if (r >= total) return;
    float re[8] = {0.f, 0.f, 0.f, 0.f, 0.f, 0.f, 0.f, 0.f};
    float im[8] = {0.f, 0.f, 0.f, 0.f, 0.f, 0.f, 0.f, 0.f};
    const float* p = h + (long long)r * GT;
#pragma unroll
    for (int t = 0; t < GT; ++t) {
        float v = p[t];
#pragma unroll
        for (int k = 0; k < 8; ++k) { re[k] += v * tc[t * 8 + k]; im[k] += v * ts[t * 8 + k]; }
    }
    float* o = ft1 + (long long)r * 16;
#pragma unroll
    for (int k = 0; k < 8; ++k) { o[k] = re[k]; o[8 + k] = im[k]; }
}

// ---------------------------------------------------------------------------
// Forward partial DFT along Y: 128 -> Q=2*m2 modes (complex->complex)
// ---------------------------------------------------------------------------
__global__ void k_dft_y(const float* __restrict__ ft1, float* __restrict__ ft2,
                        int m2, int total) {
    __shared__ float tc[24 * 128], ts[24 * 128];
    int Q = 2 * m2;
    for (int i = threadIdx.x; i < Q * 128; i += blockDim.x) {
        int q = i >> 7, y = i & 127;
        int ky = (q < m2) ? q : (GY - 2 * m2 + q);
        float ang = -TWO_PI * (float)(ky * y) / (float)GY;
        float s, c; __sincosf(ang, &s, &c);
        tc[i] = c; ts[i] = s;
    }
    __syncthreads();
    int idx = blockIdx.x * blockDim.x + threadIdx.x;
    if (idx >= total) return;
    int kz  = idx & 7;
    int q   = (idx >> 3) % Q;
    int row = idx / (8 * Q);         // row = bc*GX + x
    float ar = 0.f, ai = 0.f;
    const float* tcq = tc + q * 128;
    const float* tsq = ts + q * 128;
    for (int y = 0; y < GY; ++y) {
        int j = (row * GY + y) * 16 + kz;
        float zr = ft1[j], zi = ft1[j + 8];
        float c = tcq[y], s = tsq[y];
        ar += zr * c - zi * s;
        ai += zr * s + zi * c;
    }
    int o = (row * Q + q) * 16 + kz;
    ft2[o] = ar; ft2[o + 8] = ai;
}

// ---------------------------------------------------------------------------
// Forward partial DFT along X: 128 -> P=2*m1 modes
// ---------------------------------------------------------------------------
__global__ void k_dft_x(const float* __restrict__ ft2, float* __restrict__ ft3,
                        int m1, int m2, int total) {
    __shared__ float tc[24 * 128], ts[24 * 128];
    int P = 2 * m1, Q = 2 * m2;
    for (int i = threadIdx.x; i < P * 128; i += blockDim.x) {
        int p = i >> 7, xx = i & 127;
        int kx = (p < m1) ? p : (GX - 2 * m1 + p);
        float ang = -TWO_PI * (float)(kx * xx) / (float)GX;
        float s, c; __sincosf(ang, &s, &c);
        tc[i] = c; ts[i] = s;
    }
    __syncthreads();
    int idx = blockIdx.x * blockDim.x + threadIdx.x;
    if (idx >= total) return;
    int kz = idx & 7;
    int q  = (idx >> 3) % Q;
    int p  = (idx / (8 * Q)) % P;
    int bc = idx / (8 * Q * P);
    float ar = 0.f, ai = 0.f;
    const float* tcp = tc + p * 128;
    const float* tsp = ts + p * 128;
    for (int x = 0; x < GX; ++x) {
        int j = ((bc * GX + x) * Q + q) * 16 + kz;
        float zr = ft2[j], zi = ft2[j + 8];
        float c = tcp[x], s = tsp[x];
        ar += zr * c - zi * s;
        ai += zr * s + zi * c;
    }
    int o = ((bc * P + p) * Q + q) * 16 + kz;
    ft3[o] = ar; ft3[o + 8] = ai;
}

// ---------------------------------------------------------------------------
// Mode-space channel mixing (complex einsum over i), 4 corners by index calc.
// sw layout (4,2,Ci,Co,m1,m2,8)
// ---------------------------------------------------------------------------
__global__ void k_einsum(const float* __restrict__ ft3, const float* __restrict__ sw,
                         float* __restrict__ g, int Ci, int Co, int m1, int m2, int total) {
    int idx = blockIdx.x * blockDim.x + threadIdx.x;
    if (idx >= total) return;
    int P = 2 * m1, Q = 2 * m2;
    int kz = idx & 7;
    int q  = (idx >> 3) % Q;
    int p  = (idx / (8 * Q)) % P;
    int o  = (idx / (8 * Q * P)) % Co;
    int b  = idx / (8 * Q * P * Co);
    int c2 = (p >= m1 ? 1 : 0) + (q >= m2 ? 2 : 0);
    int aa = (p < m1) ? p : p - m1;
    int bb = (q < m2) ? q : q - m2;
    int wstride = Co * m1 * m2 * 8;                     // i-stride
    // ((((c2*2 + reim)*Ci + i)*Co + o)*m1 + aa)*(m2*8) + bb*8 + kz
    int wre = ((((c2 * 2 + 0) * Ci + 0) * Co + o) * m1 + aa) * (m2 * 8) + bb * 8 + kz;
    int wim = wre + Ci * wstride;
    int zidx = (((b * Ci + 0) * P + p) * Q + q) * 16 + kz;
    int zstride = P * Q * 16;
    float ar = 0.f, ai = 0.f;
    for (int i = 0; i < Ci; ++i) {
        float zr = ft3[zidx], zi = ft3[zidx + 8];
        float wr = sw[wre],   wi = sw[wim];
        ar += zr * wr - zi * wi;
        ai += zr * wi + zi * wr;
        zidx += zstride; wre += wstride; wim += wstride;
    }
    int oo = (((b * Co + o) * P + p) * Q + q) * 16 + kz;
    g[oo] = ar; g[oo + 8] = ai;
}

// ---------------------------------------------------------------------------
// Inverse DFT along X: P modes -> 128 positions
// ---------------------------------------------------------------------------
__global__ void k_inv_x(const float* __restrict__ g, float* __restrict__ u1,
                        int m1, int m2, int total) {
    __shared__ float tc[24 * 128], ts[24 * 128];
    int P = 2 * m1, Q = 2 * m2;
    for (int i = threadIdx.x; i < P * 128; i += blockDim.x) {
        int p = i >> 7, xx = i & 127;
        int kx = (p < m1) ? p : (GX - 2 * m1 + p);
        float ang = TWO_PI * (float)(kx * xx) / (float)GX;
        float s, c; __sincosf(ang, &s, &c);
        tc[i] = c; ts[i] = s;
    }
    __syncthreads();
    int idx = blockIdx.x * blockDim.x + threadIdx.x;
    if (idx >= total) return;
    int kz = idx & 7;
    int q  = (idx >> 3) % Q;
    int x  = (idx / (8 * Q)) % GX;
    int bo = idx / (8 * Q * GX);
    float ar = 0.f, ai = 0.f;
    for (int p = 0; p < P; ++p) {
        int j = ((bo * P + p) * Q + q) * 16 + kz;
        float zr = g[j], zi = g[j + 8];
        float c = tc[p * 128 + x], s = ts[p * 128 + x];
        ar += zr * c - zi * s;
        ai += zr * s + zi * c;
    }
    int o = ((bo * GX + x) * Q + q) * 16 + kz;
    u1[o] = ar; u1[o + 8] = ai;
}

// ---------------------------------------------------------------------------
// Inverse DFT along Y: Q modes -> 128 positions
// ---------------------------------------------------------------------------
__global__ void k_inv_y(const float* __restrict__ u1, float* __restrict__ u2,
                        int m2, int total) {
    __shared__ float tc[24 * 128], ts[24 * 128];
    int Q = 2 * m2;
    for (int i = threadIdx.x; i < Q * 128; i += blockDim.x) {
        int q = i >> 7, y = i & 127;
        int ky = (q < m2) ? q : (GY - 2 * m2 + q);
        float ang = TWO_PI * (float)(ky * y) / (float)GY;
        float s, c; __sincosf(ang, &s, &c);
        tc[i] = c; ts[i] = s;
    }
    __syncthreads();
    int idx = blockIdx.x * blockDim.x + threadIdx.x;
    if (idx >= total) return;
    int kz  = idx & 7;
    int y   = (idx >> 3) % GY;
    int box = idx / (8 * GY);          // box = bo*GX + x
    float ar = 0.f, ai = 0.f;
    for (int q = 0; q < Q; ++q) {
        int j = (box * Q + q) * 16 + kz;
        float zr = u1[j], zi = u1[j + 8];
        float c = tc[q * 128 + y], s = ts[q * 128 + y];
        ar += zr * c - zi * s;
        ai += zr * s + zi * c;
    }
    int o = (box * GY + y) * 16 + kz;
    u2[o] = ar; u2[o + 8] = ai;
}

// ---------------------------------------------------------------------------
// Inverse real DFT along T (irfft with bins 8..10 == 0) + 1/(X*Y*T) scale.
// ---------------------------------------------------------------------------
__global__ void k_inv_t(const float* __restrict__ u2, float* __restrict__ x1, int total) {
    __shared__ float tc[8 * GT], ts[8 * GT];
    for (int i = threadIdx.x; i < 8 * GT; i += blockDim.x) {
        int z = i / GT, t = i % GT;
        float ang = TWO_PI * (float)(z * t) / (float)GT;
        float s, c; __sincosf(ang, &s, &c);
        tc[i] = c; ts[i] = s;
    }
    __syncthreads();
    int r = blockIdx.x * blockDim.x + threadIdx.x;
    if (r >= total) return;
    float re[8], im[8];
#pragma unroll
    for (int k = 0; k < 8; ++k) { re[k] = u2[r * 16 + k]; im[k] = u2[r * 16 + 8 + k]; }
    const float sc = 1.0f / ((float)GX * (float)GY * (float)GT);
#pragma unroll
    for (int t = 0; t < GT; ++t) {
        float v = re[0];
#pragma unroll
        for (int z = 1; z < 8; ++z)
            v += 2.0f * (re[z] * tc[z * GT + t] - im[z] * ts[z * GT + t]);
        x1[(long long)r * GT + t] = v * sc;
    }
}

// ---------------------------------------------------------------------------
// Fused pointwise conv + residual + bias (+ gelu) with f32 WMMA 16x16x4.
// One wave per (b, 16-point tile); computes BOTH 16-wide N-tiles (Co<=32)
// from one set of A fragments. B-fragment OOB handled by clamped-index load
// + float mask (no EXEC divergence -> clean back-to-back WMMA issue).
// ---------------------------------------------------------------------------
template<int CI, int CO, int ACT>
__global__ void k_combine_wmma(const float* __restrict__ hc, const float* __restrict__ cw,
                               const float* __restrict__ cb, float* __restrict__ ha) {
    int wave = blockIdx.x * (blockDim.x >> 5) + (threadIdx.x >> 5);
    int lane = threadIdx.x & 31;
    int ln = lane & 15, hi = lane >> 4;
    const int MT = S / 16;                 // 20480 tiles per batch
    int mt = wave % MT;
    int b  = wave / MT;
    int pt0 = mt * 16;

    const int n0 = ln;                     // first N-tile: always < CO (CO>=24)
    const int n1 = 16 + ln;                // second N-tile: may exceed CO
    const int n1c = (n1 < CO) ? n1 : (CO - 1);
    const float msk1 = (n1 < CO) ? 1.f : 0.f;

    int base0 = (b * CO + n0) * S + pt0 + 8 * hi;
    int base1 = (b * CO + n1c) * S + pt0 + 8 * hi;
    __builtin_prefetch(ha + base0, 0, 3);   // x1 rows consumed in epilogue
    __builtin_prefetch(ha + base1, 0, 3);   // -> global_prefetch_b8

    v8f acc0 = {0.f, 0.f, 0.f, 0.f, 0.f, 0.f, 0.f, 0.f};
    v8f acc1 = {0.f, 0.f, 0.f, 0.f, 0.f, 0.f, 0.f, 0.f};
#pragma unroll
    for (int ks = 0; ks < (CI >> 2); ++ks) {
        const int k0 = ks * 4 + 2 * hi;
        v2f av, bv0, bv1;
        av.x = hc[(b * CI + k0)     * S + pt0 + ln];
        av.y = hc[(b * CI + k0 + 1) * S + pt0 + ln];
        bv0.x = cw[n0 * CI + k0];
        bv0.y = cw[n0 * CI + k0 + 1];
        bv1.x = cw[n1c * CI + k0]     * msk1;
        bv1.y = cw[n1c * CI + k0 + 1] * msk1;
        acc0 = __builtin_amdgcn_wmma_f32_16x16x4_f32(false, av, false, bv0,
                                                     (short)0, acc0, false, false);
        acc1 = __builtin_amdgcn_wmma_f32_16x16x4_f32(false, av, false, bv1,
                                                     (short)0, acc1, false, false);
    }
    {
        float bias = cb[n0];
#pragma unroll
        for (int r = 0; r < 8; ++r) {
            float v = acc0[r] + bias + ha[base0 + r];
            if (ACT) v = gelu_exact(v);
            ha[base0 + r] = v;
        }
    }
    if (n1 < CO) {
        float bias = cb[n1];
#pragma unroll
        for (int r = 0; r < 8; ++r) {
            float v = acc1[r] + bias + ha[base1 + r];
            if (ACT) v = gelu_exact(v);
            ha[base1 + r] = v;
        }
    }
}

// ---------------------------------------------------------------------------
// Fused fc1 (32->128, WMMA) + gelu + fc2 (128->3) through LDS.
// One wave per 16-point tile; blockDim = 128 (4 waves).
// ---------------------------------------------------------------------------
__global__ void k_fc1fc2(const float* __restrict__ h, const float* __restrict__ w1,
                         const float* __restrict__ b1, const float* __restrict__ w2,
                         const float* __restrict__ b2, float* __restrict__ out) {
    __shared__ float lds[4 * 16 * 128];
    int wv   = threadIdx.x >> 5;
    int wave = blockIdx.x * 4 + wv;
    int lane = threadIdx.x & 31;
    int ln = lane & 15, hi = lane >> 4;
    const int MT = S / 16;
    int mt = wave % MT;
    int b  = wave / MT;
    int pt0 = mt * 16;
    v2f a[8];
#pragma unroll
    for (int ks = 0; ks < 8; ++ks) {
        int k0 = ks * 4 + 2 * hi;
        a[ks].x = h[(b * 32 + k0)     * S + pt0 + ln];
        a[ks].y = h[(b * 32 + k0 + 1) * S + pt0 + ln];
    }
    float* myl = lds + wv * 16 * 128;
#pragma unroll
    for (int ntile = 0; ntile < 8; ++ntile) {
        v8f acc = {0.f, 0.f, 0.f, 0.f, 0.f, 0.f, 0.f, 0.f};
        int j = ntile * 16 + ln;
#pragma unroll
        for (int ks = 0; ks < 8; ++ks) {
            int k0 = ks * 4 + 2 * hi;
            v2f bv;
            bv.x = w1[k0 * 128 + j];
            bv.y = w1[(k0 + 1) * 128 + j];
            acc = __builtin_amdgcn_wmma_f32_16x16x4_f32(false, a[ks], false, bv,
                                                        (short)0, acc, false, false);
        }
        float bj = b1[j];
#pragma unroll
        for (int r = 0; r < 8; ++r) {
            int m = r + 8 * hi;
            myl[m * 128 + j] = gelu_exact(acc[r] + bj);
        }
    }
    __syncthreads();
    for (int t = lane; t < 48; t += 32) {
        int m = t / 3, d = t % 3;
        float acc2 = b2[d];
        for (int j = 0; j < 128; ++j) acc2 += myl[m * 128 + j] * w2[j * 3 + d];
        out[((long long)b * S + pt0 + m) * 3 + d] = acc2;
    }
}

// ---------------------------------------------------------------------------
extern "C" void kernel_launch(void* const* d_in, const int* in_sizes, int n_in,
                              void* d_out, int out_size, void* d_ws, size_t ws_size,
                              hipStream_t stream) {
    (void)in_sizes; (void)n_in; (void)out_size; (void)ws_size;
    const float* x    = (const float*)d_in[0];
    const float* fc0w = (const float*)d_in[1];
    const float* fc0b = (const float*)d_in[2];
    const float* sw[4]; const float* cwp[4]; const float* cbp[4];
    for (int i = 0; i < 4; ++i) {
        sw[i]  = (const float*)d_in[3 + 3 * i];
        cwp[i] = (const float*)d_in[4 + 3 * i];
        cbp[i] = (const float*)d_in[5 + 3 * i];
    }
    const float* fc1w = (const float*)d_in[15];
    const float* fc1b = (const float*)d_in[16];
    const float* fc2w = (const float*)d_in[17];
    const float* fc2b = (const float*)d_in[18];
    float* out = (float*)d_out;

    // workspace partition (floats); u2 aliases ft1, u1 aliases ft2
    float* W = (float*)d_ws;
    const size_t HSZ = (size_t)BATCH * 32 * S;            // 41,943,040
    const size_t FT1 = (size_t)BATCH * 32 * XY * 16;      // 33,554,432
    const size_t FT2 = (size_t)BATCH * 32 * GX * 24 * 16; // 6,291,456
    const size_t FT3 = (size_t)BATCH * 32 * 24 * 24 * 16; // 1,179,648
    float* hA   = W;
    float* hB   = hA + HSZ;
    float* ft1  = hB + HSZ;
    float* ft2  = ft1 + FT1;
    float* ft3  = ft2 + FT2;
    float* gbuf = ft3 + FT3;

    const int Ci_[4] = {20, 24, 24, 32};
    const int Co_[4] = {24, 24, 32, 32};
    const int M1_[4] = {8, 8, 12, 12};
    const int M2_[4] = {8, 8, 12, 12};

    {
        int total = BATCH * S;
        k_fc0<<<(total + 255) / 256, 256, 0, stream>>>(x, fc0w, fc0b, hA, total);
    }
    float* hc = hA;
    float* ha = hB;
    for (int L = 0; L < 4; ++L) {
        int Ci = Ci_[L], Co = Co_[L], m1 = M1_[L], m2 = M2_[L];
        int P = 2 * m1, Q = 2 * m2;
        {
            int total = BATCH * Ci * XY;
            k_dft_t<<<(total + 255) / 256, 256, 0, stream>>>(hc, ft1, total);
        }
        {
            int total = BATCH * Ci * GX * Q * 8;
            k_dft_y<<<(total + 255) / 256, 256, 0, stream>>>(ft1, ft2, m2, total);
        }
        {
            int total = BATCH * Ci * P * Q * 8;
            k_dft_x<<<(total + 255) / 256, 256, 0, stream>>>(ft2, ft3, m1, m2, total);
        }
        {
            int total = BATCH * Co * P * Q * 8;
            k_einsum<<<(total + 255) / 256, 256, 0, stream>>>(ft3, sw[L], gbuf, Ci, Co, m1, m2, total);
        }
        {
            int total = BATCH * Co * GX * Q * 8;
            k_inv_x<<<(total + 255) / 256, 256, 0, stream>>>(gbuf, ft2 /*u1*/, m1, m2, total);
        }
        {
            int total = BATCH * Co * GX * GY * 8;
            k_inv_y<<<(total + 255) / 256, 256, 0, stream>>>(ft2 /*u1*/, ft1 /*u2*/, m2, total);
        }
        {
            int total = BATCH * Co * XY;
            k_inv_t<<<(total + 255) / 256, 256, 0, stream>>>(ft1 /*u2*/, ha, total);
        }
        {
            int waves  = BATCH * (S / 16);       // one wave per 16-pt tile (2 n-tiles inside)
            int blocks = waves / 8;              // 8 waves per 256-thread block
            switch (L) {
                case 0: k_combine_wmma<20, 24, 1><<<blocks, 256, 0, stream>>>(hc, cwp[L], cbp[L], ha); break;
                case 1: k_combine_wmma<24, 24, 1><<<blocks, 256, 0, stream>>>(hc, cwp[L], cbp[L], ha); break;
                case 2: k_combine_wmma<24, 32, 1><<<blocks, 256, 0, stream>>>(hc, cwp[L], cbp[L], ha); break;
                default: k_combine_wmma<32, 32, 0><<<blocks, 256, 0, stream>>>(hc, cwp[L], cbp[L], ha); break;
            }
        }
        float* tmp = hc; hc = ha; ha = tmp;
    }
    {
        int blocks = (BATCH * (S / 16)) / 4;     // 4 waves (tiles) per block
        k_fc1fc2<<<blocks, 128, 0, stream>>>(hc, fc1w, fc1b, fc2w, fc2b, out);
    }
}